// ODEINDLayer_89713276879370
// MI455X (gfx1250) — compile-verified
//
#include <hip/hip_runtime.h>
#include <cmath>

// Static problem config (matches the reference's init_kwargs)
#define B_TOTAL 1024      // BS*NIND
#define S_STEPS 64
#define T_STEPS 63
#define NB      192       // N = S*D, D = 3
#define BW      6         // half bandwidth of AtA (central rows couple +-2 blocks)
#define BD      7         // band diagonals stored (0..6)
#define LANES   32        // wave32: one batch system per lane

// Lane-interleaved LDS accessors (conflict-free: consecutive lanes hit consecutive banks)
#define BANDI(r,d) s_band[((r)*BD + (d))*LANES + lane]
#define YI(r)      s_y[(r)*LANES + lane]
#define STP(t)     s_steps[lane*T_STEPS + (t)]

typedef unsigned uint32x4 __attribute__((ext_vector_type(4)));
typedef int      int32x4  __attribute__((ext_vector_type(4)));
typedef int      int32x8  __attribute__((ext_vector_type(8)));

// TDM group-1 flag words (D# per CDNA5 ISA ch.8):
//   data_size=2 (4 bytes)                                -> bits[17:16]
//   pad_enable                                           -> bit[20]
//   pad_interval code 5 = every 64 DWORDs                -> bits[24:22]
//   pad_amount  code 0 = 1 DWORD                         -> bits[31:25]
// Padded LDS element offset becomes n + n/64 -> row strides 195/65, conflict-free banks.
#define TDM_FLAGS_PLAIN  (2u << 16)
#define TDM_FLAGS_PAD64  ((2u << 16) | (1u << 20) | (5u << 22))

// Build a 2-D TDM descriptor and fire it. Tile = tile1 rows x tile0 f32 elements,
// global row stride = stride0 elements. LDS side is streamed contiguously
// (plus optional hardware padding on loads).
__device__ __forceinline__ void tdm_2d(bool is_store, unsigned flags0, unsigned lds_off,
                                       unsigned long long gaddr,
                                       unsigned dim0, unsigned dim1,
                                       unsigned tile0, unsigned tile1,
                                       unsigned stride0)
{
    uint32x4 g0;
    g0.x = 1u;                                              // count = 1 (valid D#)
    g0.y = lds_off;                                         // lds_addr
    g0.z = (unsigned)(gaddr & 0xffffffffull);               // global_addr[31:0]
    g0.w = (unsigned)((gaddr >> 32) & 0x1ffffffull)         // global_addr[56:32]
         | (2u << 30);                                      // type = 2 ("image")
    int32x8 g1;
    g1[0] = (int)flags0;                                    // mask=0 | data_size | pad ctl
    g1[1] = (int)(dim0 << 16);                              // tensor_dim0[15:0]
    g1[2] = (int)((dim0 >> 16) | (dim1 << 16));             // dim0[31:16] | dim1[15:0]
    g1[3] = (int)((dim1 >> 16) | (tile0 << 16));            // dim1[31:16] | tile_dim0
    g1[4] = (int)tile1;                                     // tile_dim1 | tile_dim2=0
    g1[5] = (int)stride0;                                   // tensor_dim0_stride[31:0]
    g1[6] = 0;
    g1[7] = 0;
    int32x4 z4 = {0, 0, 0, 0};
#if defined(__clang_major__) && (__clang_major__ >= 23)
    int32x8 z8 = {0, 0, 0, 0, 0, 0, 0, 0};
    if (is_store) __builtin_amdgcn_tensor_store_from_lds(g0, g1, z4, z4, z8, 0);
    else          __builtin_amdgcn_tensor_load_to_lds  (g0, g1, z4, z4, z8, 0);
#else
    if (is_store) __builtin_amdgcn_tensor_store_from_lds(g0, g1, z4, z4, 0);
    else          __builtin_amdgcn_tensor_load_to_lds  (g0, g1, z4, z4, 0);
#endif
}

__global__ __launch_bounds__(LANES)
void ode_band_solve(const float* __restrict__ coeffs,   // (B, 64, 1, 1, 3)
                    const float* __restrict__ rhs,      // (B, 64, 1)
                    const float* __restrict__ iv_rhs,   // (B, 2, 1, 2)
                    const float* __restrict__ steps,    // (B, 63)
                    float* __restrict__ out)            // (B, 64, 1, 3)
{
    __shared__ float s_band[NB * BD * LANES];          // 172032 B : AtA band -> L band
    __shared__ float s_y[NB * LANES];                  //  24576 B : rhs -> y -> x (in place)
    __shared__ float s_steps[T_STEPS * LANES];         //   8064 B : steps, [lane][t]
    __shared__ float s_xout[NB * LANES];               //  24576 B : batch-major TDM-store staging
    __shared__ float s_cf[NB * LANES + NB*LANES/64];   //  24960 B : coeffs, TDM hw-padded (stride 195)
    __shared__ float s_rv[S_STEPS * LANES + LANES];    //   8320 B : rhs,    TDM hw-padded (stride 65)

    const int lane = threadIdx.x;
    const int b0   = blockIdx.x * LANES;
    const int b    = b0 + lane;

    // ---- three TDM descriptors stage all bulk inputs; DMA overlaps band zeroing.
    //      TDM ops complete in-order, so tensorcnt<=2 means steps is resident. ----
    tdm_2d(false, TDM_FLAGS_PLAIN, (unsigned)(uintptr_t)(&s_steps[0]),
           (unsigned long long)(uintptr_t)steps + (unsigned long long)b0 * T_STEPS * 4ull,
           T_STEPS, LANES, T_STEPS, LANES, T_STEPS);
    tdm_2d(false, TDM_FLAGS_PAD64, (unsigned)(uintptr_t)(&s_cf[0]),
           (unsigned long long)(uintptr_t)coeffs + (unsigned long long)b0 * NB * 4ull,
           NB, LANES, NB, LANES, NB);
    tdm_2d(false, TDM_FLAGS_PAD64, (unsigned)(uintptr_t)(&s_rv[0]),
           (unsigned long long)(uintptr_t)rhs + (unsigned long long)b0 * S_STEPS * 4ull,
           S_STEPS, LANES, S_STEPS, LANES, S_STEPS);

    const float* iv = iv_rhs + (size_t)b * 4;          // 4 floats / batch (tiny: direct)
    __builtin_prefetch(iv, 0, 3);                      // global_prefetch_b8

    // Zero the band while the tensor DMAs are in flight
    for (int i = 0; i < NB * BD; ++i) s_band[i * LANES + lane] = 0.0f;

    __builtin_amdgcn_s_wait_tensorcnt((short)2);       // steps tile resident
    asm volatile("" ::: "memory");

    // ---- analytic band assembly of AtA = A^T A (smoothness rows) ----
    for (int st = 0; st < T_STEPS; ++st) {
        const float s  = STP(st);
        const float s2 = s * s;
        const int   r0 = 3 * st;
        {   // forward, i=0: cols {r0,r0+1,r0+2,r0+3}, vals {1, s, s2/2, -1}
            const float v0 = 1.f, v1 = s, v2 = 0.5f * s2, v3 = -1.f;
            BANDI(r0,0)   += v0*v0;
            BANDI(r0+1,1) += v1*v0;  BANDI(r0+1,0) += v1*v1;
            BANDI(r0+2,2) += v2*v0;  BANDI(r0+2,1) += v2*v1;  BANDI(r0+2,0) += v2*v2;
            BANDI(r0+3,3) += v3*v0;  BANDI(r0+3,2) += v3*v1;  BANDI(r0+3,1) += v3*v2;  BANDI(r0+3,0) += v3*v3;
        }
        {   // forward, i=1: cols {r0+1,r0+2,r0+4}, vals {s, s2, -s}
            const float v0 = s, v1 = s2, v2 = -s;
            BANDI(r0+1,0) += v0*v0;
            BANDI(r0+2,1) += v1*v0;  BANDI(r0+2,0) += v1*v1;
            BANDI(r0+4,3) += v2*v0;  BANDI(r0+4,2) += v2*v1;  BANDI(r0+4,0) += v2*v2;
        }
        {   // backward, i=0: cols {r0,r0+3,r0+4,r0+5}, vals {-1, 1, -s, s2/2}
            const float v0 = -1.f, v1 = 1.f, v2 = -s, v3 = 0.5f * s2;
            BANDI(r0,0)   += v0*v0;
            BANDI(r0+3,3) += v1*v0;  BANDI(r0+3,0) += v1*v1;
            BANDI(r0+4,4) += v2*v0;  BANDI(r0+4,1) += v2*v1;  BANDI(r0+4,0) += v2*v2;
            BANDI(r0+5,5) += v3*v0;  BANDI(r0+5,2) += v3*v1;  BANDI(r0+5,1) += v3*v2;  BANDI(r0+5,0) += v3*v3;
        }
        {   // backward, i=1: cols {r0+1,r0+4,r0+5}, vals {s, -s, s2}
            const float v0 = s, v1 = -s, v2 = s2;
            BANDI(r0+1,0) += v0*v0;
            BANDI(r0+4,3) += v1*v0;  BANDI(r0+4,0) += v1*v1;
            BANDI(r0+5,4) += v2*v0;  BANDI(r0+5,1) += v2*v1;  BANDI(r0+5,0) += v2*v2;
        }
    }
    for (int st = 1; st < T_STEPS; ++st) {  // central rows, st = 1..62
        const float cp = STP(st) + STP(st - 1);
        const int c0 = 3*st - 2, c1 = 3*st + 2, c2 = 3*st + 4;  // diffs 4, 2, 6
        const float v0 = -1.f, v1 = -cp, v2 = 1.f;
        BANDI(c0,0) += v0*v0;
        BANDI(c1,4) += v1*v0;  BANDI(c1,0) += v1*v1;
        BANDI(c2,6) += v2*v0;  BANDI(c2,2) += v2*v1;  BANDI(c2,0) += v2*v2;
    }

    __builtin_amdgcn_s_wait_tensorcnt((short)0);       // coeffs + rhs tiles resident
    asm volatile("" ::: "memory");

    // ---- equation blocks: AetAe = a a^T on the diagonal, Aetbe = a*rhs ----
    // Hardware-padded layouts: coeff (lane,x) at 195*lane + x + (x>>6); rhs at 65*lane + st.
    {
        const int cbase = lane * (NB + NB/64);         // 195
        const int rbase = lane * (S_STEPS + 1);        // 65
        for (int st = 0; st < S_STEPS; ++st) {
            const int x = 3 * st;
            const float a0 = s_cf[cbase + x     + ( x      >> 6)];
            const float a1 = s_cf[cbase + x + 1 + ((x + 1) >> 6)];
            const float a2 = s_cf[cbase + x + 2 + ((x + 2) >> 6)];
            const int r0 = 3 * st;
            BANDI(r0,0)   += a0*a0;
            BANDI(r0+1,1) += a1*a0;  BANDI(r0+1,0) += a1*a1;
            BANDI(r0+2,2) += a2*a0;  BANDI(r0+2,1) += a2*a1;  BANDI(r0+2,0) += a2*a2;
            const float re = s_rv[rbase + st];
            YI(r0)   = a0 * re;
            YI(r0+1) = a1 * re;
            YI(r0+2) = a2 * re;
        }
    }
    // initial-value regularization on first 2 step blocks (INIT_IDX = {0,1})
#pragma unroll
    for (int st = 0; st < 2; ++st) {
        BANDI(3*st,0)   += 1.0f;
        BANDI(3*st+1,0) += 1.0f;
        YI(3*st)   += iv[st*2 + 0];
        YI(3*st+1) += iv[st*2 + 1];
    }

    // ---- banded Cholesky (in-place on the band) ----
    for (int r = 0; r < NB; ++r) {
        const int jmin = (r > BW) ? (r - BW) : 0;
        for (int j = jmin; j < r; ++j) {
            float sum = BANDI(r, r - j);
            for (int k = jmin; k < j; ++k)
                sum -= BANDI(r, r - k) * BANDI(j, j - k);
            BANDI(r, r - j) = sum / BANDI(j, 0);
        }
        float sum = BANDI(r, 0);
        for (int k = jmin; k < r; ++k) {
            const float l = BANDI(r, r - k);
            sum -= l * l;
        }
        BANDI(r, 0) = sqrtf(sum);
    }

    // ---- forward substitution: L y = b (in place) ----
    for (int r = 0; r < NB; ++r) {
        const int jmin = (r > BW) ? (r - BW) : 0;
        float sum = YI(r);
        for (int j = jmin; j < r; ++j) sum -= BANDI(r, r - j) * YI(j);
        YI(r) = sum / BANDI(r, 0);
    }
    // ---- backward substitution: L^T x = y (in place) ----
    for (int r = NB - 1; r >= 0; --r) {
        const int jmax = (r + BW < NB - 1) ? (r + BW) : (NB - 1);
        float sum = YI(r);
        for (int j = r + 1; j <= jmax; ++j) sum -= BANDI(j, j - r) * YI(j);
        YI(r) = sum / BANDI(r, 0);
    }

    // ---- re-layout x into batch-major staging (rotation schedule keeps both the
    //      interleaved reads and the contiguous writes bank-conflict-free) ----
    for (int i = 0; i < NB; ++i) {
        int r = i + lane; if (r >= NB) r -= NB;        // r = (i + lane) mod 192
        s_xout[lane * NB + r] = YI(r);
    }
    asm volatile("s_wait_dscnt 0" ::: "memory");       // LDS writes visible to TDM

    // ---- one TDM descriptor stores the 32x192 solution tile, fully coalesced;
    //      S_ENDPGM's implicit wait-idle drains TENSORcnt ----
    tdm_2d(true, TDM_FLAGS_PLAIN, (unsigned)(uintptr_t)(&s_xout[0]),
           (unsigned long long)(uintptr_t)out + (unsigned long long)b0 * NB * 4ull,
           NB, LANES, NB, LANES, NB);
}

extern "C" void kernel_launch(void* const* d_in, const int* in_sizes, int n_in,
                              void* d_out, int out_size, void* d_ws, size_t ws_size,
                              hipStream_t stream) {
    (void)in_sizes; (void)n_in; (void)out_size; (void)d_ws; (void)ws_size;
    const float* coeffs = (const float*)d_in[0];
    const float* rhs    = (const float*)d_in[1];
    const float* iv_rhs = (const float*)d_in[2];
    const float* steps  = (const float*)d_in[3];
    float* out = (float*)d_out;

    dim3 grid(B_TOTAL / LANES);   // 32 workgroups
    dim3 block(LANES);            // 1 wave32 per workgroup (LDS-limited: ~262KB/WG)
    hipLaunchKernelGGL(ode_band_solve, grid, block, 0, stream,
                       coeffs, rhs, iv_rhs, steps, out);
}